// Time_Series_Tokenization_63058709840112
// MI455X (gfx1250) — compile-verified
//
#include <hip/hip_runtime.h>

// ---- problem constants (match reference) ----
#define B_  16
#define C_  64
#define L_  512
#define D_  768
#define LP_ 16
#define NP_ 68            // (512+16)/8 + 2
#define ROWS_ (B_*C_)     // 1024
#define PERIOD_ 7

typedef float v2f __attribute__((ext_vector_type(2)));
typedef float v4f __attribute__((ext_vector_type(4)));
typedef float v8f __attribute__((ext_vector_type(8)));

// -------------------------------------------------------------------------
// Kernel 1: per-(b,l) mean / unbiased var over C; m = mean * rsqrt(var)
// -------------------------------------------------------------------------
__global__ void k_stats(const float* __restrict__ x, float* __restrict__ m) {
    int idx = blockIdx.x * blockDim.x + threadIdx.x;   // 0 .. B*L-1
    if (idx >= B_ * L_) return;
    int b = idx / L_;
    int l = idx - b * L_;
    const float* xp = x + (size_t)b * C_ * L_ + l;
    float s = 0.f, s2 = 0.f;
    #pragma unroll 8
    for (int c = 0; c < C_; ++c) {
        float v = xp[(size_t)c * L_];
        s  += v;
        s2 += v * v;
    }
    float mean = s * (1.0f / C_);
    float var  = (s2 - (float)C_ * mean * mean) * (1.0f / (C_ - 1));
    m[idx] = mean * __frsqrt_rn(var);
}

// -------------------------------------------------------------------------
// Kernel 2: per-row affine + seasonal decomposition (period 7).
// One 128-thread block per row; xt staged in LDS; per-phase sums via
// LDS float atomics (ds_add_f32).
// -------------------------------------------------------------------------
__global__ void k_decompose(const float* __restrict__ x,
                            const float* __restrict__ gamma,
                            const float* __restrict__ beta,
                            const float* __restrict__ m,
                            float* __restrict__ tr_o,
                            float* __restrict__ se_o,
                            float* __restrict__ re_o) {
    __shared__ float sx[L_];
    __shared__ float ssum[PERIOD_];
    __shared__ float spavg[PERIOD_];

    const int row = blockIdx.x;          // 0..1023
    const int b   = row / C_;
    const int tid = threadIdx.x;         // 0..127
    const size_t base  = (size_t)row * L_;
    const size_t mbase = (size_t)b * L_;

    if (tid < PERIOD_) ssum[tid] = 0.0f;

    // xt = gamma*(x - m) + beta  -> LDS
    #pragma unroll
    for (int j = 0; j < 4; ++j) {
        int i = tid + j * 128;
        float xv = x[base + i];
        sx[i] = gamma[base + i] * (xv - m[mbase + i]) + beta[base + i];
    }
    __syncthreads();

    // centered 7-pt moving average (valid interior 3..508), detrend, phase sums
    float trv[4];
    #pragma unroll
    for (int j = 0; j < 4; ++j) {
        int i = tid + j * 128;
        bool valid = (i >= 3) && (i < L_ - 3);
        float t = 0.0f;
        if (valid) {
            float s = 0.0f;
            #pragma unroll
            for (int k = -3; k <= 3; ++k) s += sx[i + k];
            t = s * (1.0f / 7.0f);
        }
        trv[j] = t;
        tr_o[base + i] = t;
        if (valid) {
            float det = sx[i] - t;
            atomicAdd(&ssum[i % PERIOD_], det);   // ds_add_f32
        }
    }
    __syncthreads();

    // de-meaned per-phase averages (7 values; single thread is fine)
    if (tid == 0) {
        float pv[PERIOD_];
        float mean = 0.0f;
        #pragma unroll
        for (int p = 0; p < PERIOD_; ++p) {
            int first = (p >= 3) ? p : p + 7;           // first valid idx with i%7==p
            int cnt   = (L_ - 4 - first) / 7 + 1;       // valid idx range [3, 508]
            pv[p] = ssum[p] / (float)cnt;
            mean += pv[p];
        }
        mean *= (1.0f / 7.0f);
        #pragma unroll
        for (int p = 0; p < PERIOD_; ++p) spavg[p] = pv[p] - mean;
    }
    __syncthreads();

    // seasonal (full length) and resid (interior only)
    #pragma unroll
    for (int j = 0; j < 4; ++j) {
        int i = tid + j * 128;
        bool valid = (i >= 3) && (i < L_ - 3);
        float se = spavg[i % PERIOD_];
        se_o[base + i] = se;
        re_o[base + i] = valid ? (sx[i] - trv[j] - se) : 0.0f;
    }
}

// -------------------------------------------------------------------------
// Kernel 3: three [1024,512] x [512->16] GEMMs via V_WMMA_F32_16X16X4_F32.
// grid (64 M-tiles, 3 streams), 1 wave32 per tile.
// A-frag: lanes 0-15 hold M=lane K={k0,k0+1}; lanes 16-31 hold M=lane-16,
// K={k0+2,k0+3}.  B-frag mirrored with N=lane&15.
// -------------------------------------------------------------------------
__global__ void k_gemm_patch(const float* __restrict__ tr,
                             const float* __restrict__ se,
                             const float* __restrict__ re,
                             const float* __restrict__ wt, const float* __restrict__ bt,
                             const float* __restrict__ wsn, const float* __restrict__ bs,
                             const float* __restrict__ wr, const float* __restrict__ br,
                             float* __restrict__ cat) {
    const int mt = blockIdx.x;   // 0..63
    const int s  = blockIdx.y;   // 0..2
    const float* S; const float* W; const float* bias;
    if      (s == 0) { S = tr; W = wt;  bias = bt; }
    else if (s == 1) { S = se; W = wsn; bias = bs; }
    else             { S = re; W = wr;  bias = br; }

    const int lane = threadIdx.x;
    const int hsel = lane >> 4;        // 0: K pair {0,1}, 1: K pair {2,3}
    const int mrow = lane & 15;
    const int r0   = mt * 16;

    const float* Arow = S + (size_t)(r0 + mrow) * L_ + 2 * hsel;
    const float* Brow = W + (size_t)mrow * L_ + 2 * hsel;   // W[j, k], j = lane&15

    v8f acc = {};
    #pragma unroll 4
    for (int k0 = 0; k0 < L_; k0 += 4) {
        v2f a = *(const v2f*)(Arow + k0);
        v2f b = *(const v2f*)(Brow + k0);
        acc = __builtin_amdgcn_wmma_f32_16x16x4_f32(
                  false, a, false, b, (short)0, acc, false, false);
    }

    const float bv = bias[mrow];       // bias depends on output column n = lane&15
    #pragma unroll
    for (int r = 0; r < 8; ++r) {
        int mm = r0 + r + 8 * hsel;    // C/D layout: vgpr r -> M = r + 8*(lane>=16)
        cat[(size_t)mm * 48 + s * 16 + mrow] = acc[r] + bv;
    }
}

// -------------------------------------------------------------------------
// Kernel 4: [1024,48] x [48->768] GEMM via V_WMMA_F32_16X16X4_F32.
// grid (64 M-tiles, 48 N-tiles), 1 wave32 per tile, K=48 -> 12 WMMAs.
// -------------------------------------------------------------------------
__global__ void k_gemm_embed(const float* __restrict__ cat,
                             const float* __restrict__ wg,
                             const float* __restrict__ bg,
                             float* __restrict__ p) {
    const int mt = blockIdx.x;   // 0..63
    const int nt = blockIdx.y;   // 0..47
    const int lane = threadIdx.x;
    const int hsel = lane >> 4;
    const int mrow = lane & 15;
    const int r0 = mt * 16;
    const int n0 = nt * 16;

    const float* Arow = cat + (size_t)(r0 + mrow) * 48 + 2 * hsel;
    const float* Brow = wg  + (size_t)(n0 + mrow) * 48 + 2 * hsel;  // w_g[d, k]

    v8f acc = {};
    #pragma unroll
    for (int k0 = 0; k0 < 48; k0 += 4) {
        v2f a = *(const v2f*)(Arow + k0);
        v2f b = *(const v2f*)(Brow + k0);
        acc = __builtin_amdgcn_wmma_f32_16x16x4_f32(
                  false, a, false, b, (short)0, acc, false, false);
    }

    const float bv = bg[n0 + mrow];
    #pragma unroll
    for (int r = 0; r < 8; ++r) {
        int mm = r0 + r + 8 * hsel;
        p[(size_t)mm * D_ + n0 + mrow] = acc[r] + bv;
    }
}

// -------------------------------------------------------------------------
// Kernel 5: broadcast p[row, :] to 68 patch slots; 214 MB streaming write,
// non-temporal so the output doesn't thrash L2 (p stays resident).
// grid (1024 rows, 68 slots), 192 threads, one float4 per thread.
// -------------------------------------------------------------------------
__global__ void k_broadcast(const float* __restrict__ p, float* __restrict__ out) {
    const int row = blockIdx.x;   // 0..1023
    const int np  = blockIdx.y;   // 0..67
    const int t   = threadIdx.x;  // 0..191
    const v4f* p4 = (const v4f*)(p + (size_t)row * D_);
    v4f* o4 = (v4f*)(out + ((size_t)row * NP_ + np) * D_);
    v4f v = p4[t];
    __builtin_nontemporal_store(v, o4 + t);
}

// -------------------------------------------------------------------------
extern "C" void kernel_launch(void* const* d_in, const int* in_sizes, int n_in,
                              void* d_out, int out_size, void* d_ws, size_t ws_size,
                              hipStream_t stream) {
    const float* data_x     = (const float*)d_in[0];
    // d_in[1] = data_y (unused by the reference)
    const float* gamma      = (const float*)d_in[2];
    const float* beta       = (const float*)d_in[3];
    const float* w_trend    = (const float*)d_in[4];
    const float* b_trend    = (const float*)d_in[5];
    const float* w_seasonal = (const float*)d_in[6];
    const float* b_seasonal = (const float*)d_in[7];
    const float* w_resid    = (const float*)d_in[8];
    const float* b_resid    = (const float*)d_in[9];
    const float* w_g        = (const float*)d_in[10];
    const float* b_g        = (const float*)d_in[11];
    float* out = (float*)d_out;

    // workspace layout (floats); total ~9.7 MB
    float* ws  = (float*)d_ws;
    float* m   = ws;                       // B*L            = 8192
    float* tr  = m   + B_ * L_;            // ROWS*L         = 524288
    float* se  = tr  + ROWS_ * L_;
    float* re  = se  + ROWS_ * L_;
    float* cat = re  + ROWS_ * L_;         // ROWS*48        = 49152
    float* p   = cat + ROWS_ * 48;         // ROWS*D         = 786432

    k_stats     <<<dim3((B_ * L_ + 255) / 256), dim3(256), 0, stream>>>(data_x, m);
    k_decompose <<<dim3(ROWS_),        dim3(128), 0, stream>>>(data_x, gamma, beta, m, tr, se, re);
    k_gemm_patch<<<dim3(64, 3),        dim3(32),  0, stream>>>(tr, se, re,
                                                               w_trend, b_trend,
                                                               w_seasonal, b_seasonal,
                                                               w_resid, b_resid, cat);
    k_gemm_embed<<<dim3(64, 48),       dim3(32),  0, stream>>>(cat, w_g, b_g, p);
    k_broadcast <<<dim3(ROWS_, NP_),   dim3(192), 0, stream>>>(p, out);
}